// JKNet1_55293408969100
// MI455X (gfx1250) — compile-verified
//
#include <hip/hip_runtime.h>

// JKNet GCN forward for MI455X (gfx1250, wave32).
//
// Aggregation strategy: build CSR (dst -> incoming srcs) once per call, then
// per layer do a register-accumulated gather (one wave32 per node, one float4
// per lane) instead of 102M fp32 global atomics per layer. The whole working
// set (~80 MB) is resident in the 192 MB L2, so the gather runs at L2 rate.
// GEMMs (50000x128 @ 128x128) use V_WMMA_F32_16X16X4_F32 (full fp32 matrix
// pipe), one 16x16 output tile per wave, A staged in LDS, B panel in VGPRs.

#define D 128
#define NLAYERS 4
#define BN_EPS 1e-5f
#define SCAN_B 256   // two-level scan handles N <= 256*256 = 65536 (N = 50000)

typedef __attribute__((ext_vector_type(2))) float v2f;
typedef __attribute__((ext_vector_type(8))) float v8f;

// ---------------------------------------------------------------- utilities

__global__ void zero_b32(int* __restrict__ p, long n) {
  long i = (long)blockIdx.x * blockDim.x + threadIdx.x;
  long stride = (long)gridDim.x * blockDim.x;
  for (; i < n; i += stride) p[i] = 0;
}

__global__ void degree_kernel(const int* __restrict__ src, const int* __restrict__ dst,
                              int* __restrict__ deg_out, int* __restrict__ deg_in, int E) {
  int e = blockIdx.x * blockDim.x + threadIdx.x;
  if (e < E) {
    atomicAdd(&deg_out[src[e]], 1);
    atomicAdd(&deg_in[dst[e]], 1);
  }
}

// norm[i] = rsqrt(max(deg[i], 1))
__global__ void norm_kernel(const int* __restrict__ deg, float* __restrict__ norm, int n) {
  int i = blockIdx.x * blockDim.x + threadIdx.x;
  if (i < n) norm[i] = rsqrtf(fmaxf((float)deg[i], 1.f));
}

// ------------------------------------------------------------ prefix scan (3 kernels)

__global__ void block_sum_kernel(const int* __restrict__ v, int* __restrict__ bsum, int n) {
  __shared__ int sm[SCAN_B];
  int t = threadIdx.x;
  int i = blockIdx.x * SCAN_B + t;
  sm[t] = (i < n) ? v[i] : 0;
  __syncthreads();
  for (int o = SCAN_B / 2; o > 0; o >>= 1) {
    if (t < o) sm[t] += sm[t + o];
    __syncthreads();
  }
  if (t == 0) bsum[blockIdx.x] = sm[0];
}

__global__ void scan_offsets_kernel(const int* __restrict__ bsum, int* __restrict__ boff, int nb) {
  __shared__ int sm[SCAN_B];
  int t = threadIdx.x;
  int v = (t < nb) ? bsum[t] : 0;
  sm[t] = v;
  __syncthreads();
  for (int o = 1; o < SCAN_B; o <<= 1) {
    int x = (t >= o) ? sm[t - o] : 0;
    __syncthreads();
    sm[t] += x;
    __syncthreads();
  }
  boff[t] = sm[t] - v;  // exclusive scan of block sums
}

__global__ void scan_write_kernel(const int* __restrict__ v, const int* __restrict__ boff,
                                  int* __restrict__ row_ptr, int n) {
  __shared__ int sm[SCAN_B];
  int t = threadIdx.x;
  int i = blockIdx.x * SCAN_B + t;
  int val = (i < n) ? v[i] : 0;
  sm[t] = val;
  __syncthreads();
  for (int o = 1; o < SCAN_B; o <<= 1) {
    int x = (t >= o) ? sm[t - o] : 0;
    __syncthreads();
    sm[t] += x;
    __syncthreads();
  }
  int excl = sm[t] - val + boff[blockIdx.x];
  if (i < n) row_ptr[i] = excl;
  if (i == n - 1) row_ptr[n] = excl + val;  // row_ptr[N] = E
}

__global__ void csr_fill_kernel(const int* __restrict__ src, const int* __restrict__ dst,
                                const int* __restrict__ row_ptr, int* __restrict__ fill,
                                int* __restrict__ col_idx, int E) {
  int e = blockIdx.x * blockDim.x + threadIdx.x;
  if (e < E) {
    int d = dst[e];
    int pos = atomicAdd(&fill[d], 1);
    col_idx[row_ptr[d] + pos] = src[e];
  }
}

// ----------------------------------------------------------------- gather
// One wave32 per destination node; lane L accumulates float4 L of the row in
// registers across all incoming edges, fusing both GCN norms. No atomics.
__global__ __launch_bounds__(256)
void gather_kernel(const float* __restrict__ h, const float* __restrict__ norm_src,
                   const float* __restrict__ norm_dst, const int* __restrict__ row_ptr,
                   const int* __restrict__ col_idx, float* __restrict__ agg, int N) {
  int gid = blockIdx.x * blockDim.x + threadIdx.x;
  int node = gid >> 5;            // uniform within the wave
  int lane = gid & 31;
  if (node >= N) return;
  int beg = row_ptr[node];
  int end = row_ptr[node + 1];
  float4 acc = make_float4(0.f, 0.f, 0.f, 0.f);
  for (int j = beg; j < end; ++j) {
    int s = col_idx[j];
    float ns = norm_src[s];
    float4 v = ((const float4*)(h + (long)s * D))[lane];
    acc.x = fmaf(v.x, ns, acc.x);
    acc.y = fmaf(v.y, ns, acc.y);
    acc.z = fmaf(v.z, ns, acc.z);
    acc.w = fmaf(v.w, ns, acc.w);
  }
  float nd = norm_dst[node];
  acc.x *= nd; acc.y *= nd; acc.z *= nd; acc.w *= nd;
  ((float4*)(agg + (long)node * D))[lane] = acc;
}

// ------------------------------------------------------- WMMA fp32 GEMM+ReLU
// out[m][n] = relu( in[m][:] @ W + bias[n] ); `out` may alias `in` (all A
// reads staged to LDS before the barrier; each block touches only its own 16
// rows). Requires N % 16 == 0 (50000 ok). If bnsum != nullptr, accumulates
// per-column sum / sum-of-squares of the activated output (BN batch stats).
__global__ __launch_bounds__(256)
void gemm_relu_kernel(const float* in, const float* __restrict__ W,
                      const float* __restrict__ bias, float* out,
                      float* __restrict__ bnsum, float* __restrict__ bnsumsq, int N) {
  __shared__ float As[16][132];           // pad 128->132: conflict-free rows
  int m0 = blockIdx.x * 16;

  for (int idx = threadIdx.x; idx < 16 * D; idx += 256) {
    int r = idx >> 7;
    int c = idx & (D - 1);
    As[r][c] = in[(long)(m0 + r) * D + c];
  }
  __syncthreads();

  int wave = threadIdx.x >> 5;
  int lane = threadIdx.x & 31;
  int n0 = wave * 16;
  int col = lane & 15;                    // N within tile (B/C/D layout)
  int kgrp = lane >> 4;                   // K-half selector (A/B layout)
  int arow = lane & 15;                   // M within tile (A layout)

  // Preload this wave's 128x16 panel of W into registers (64 VGPRs).
  v2f B[32];
#pragma unroll
  for (int j = 0; j < 32; ++j) {
    int k = 4 * j + 2 * kgrp;
    B[j].x = W[(long)k * D + n0 + col];
    B[j].y = W[(long)(k + 1) * D + n0 + col];
  }

  v8f acc = {};
#pragma unroll
  for (int j = 0; j < 32; ++j) {
    int k = 4 * j + 2 * kgrp;
    v2f a;
    a.x = As[arow][k];
    a.y = As[arow][k + 1];
    acc = __builtin_amdgcn_wmma_f32_16x16x4_f32(false, a, false, B[j],
                                                (short)0, acc, false, false);
  }

  float bv = bias[n0 + col];
  float s = 0.f, sq = 0.f;
#pragma unroll
  for (int i = 0; i < 8; ++i) {
    int m = i + 8 * kgrp;                 // lanes>=16 hold rows 8..15
    float v = fmaxf(acc[i] + bv, 0.f);
    out[(long)(m0 + m) * D + n0 + col] = v;
    s += v;
    sq += v * v;
  }

  if (bnsum) {
    s += __shfl_xor(s, 16, 32);           // combine half-wave column partners
    sq += __shfl_xor(sq, 16, 32);
    if (lane < 16) {
      atomicAdd(&bnsum[n0 + col], s);
      atomicAdd(&bnsumsq[n0 + col], sq);
    }
  }
}

// ----------------------------------------------------------- BN finalization
__global__ void bn_finalize(const float* __restrict__ sum, const float* __restrict__ sumsq,
                            const float* __restrict__ gamma, const float* __restrict__ beta,
                            float* __restrict__ scale, float* __restrict__ shift, float invN) {
  int c = threadIdx.x;
  float mu = sum[c] * invN;
  float var = sumsq[c] * invN - mu * mu;  // biased variance
  float inv = rsqrtf(var + BN_EPS);
  float g = gamma[c] * inv;
  scale[c] = g;
  shift[c] = beta[c] - mu * g;
}

// BN apply + ReLU + JumpingKnowledge running max.
__global__ void bn_apply(const float* __restrict__ t, const float* __restrict__ scale,
                         const float* __restrict__ shift, float* __restrict__ h,
                         float* __restrict__ jk, int first, long n) {
  long i = (long)blockIdx.x * blockDim.x + threadIdx.x;
  long stride = (long)gridDim.x * blockDim.x;
  for (; i < n; i += stride) {
    int c = (int)(i & (D - 1));
    float v = fmaxf(t[i] * scale[c] + shift[c], 0.f);
    h[i] = v;
    jk[i] = first ? v : fmaxf(jk[i], v);
  }
}

// ------------------------------------------------------------------- driver
extern "C" void kernel_launch(void* const* d_in, const int* in_sizes, int n_in,
                              void* d_out, int out_size, void* d_ws, size_t ws_size,
                              hipStream_t stream) {
  const float* x      = (const float*)d_in[0];
  const int*   src    = (const int*)d_in[1];
  const int*   dst    = (const int*)d_in[2];
  const float* Ws     = (const float*)d_in[3];
  const float* bs     = (const float*)d_in[4];
  const float* gammas = (const float*)d_in[5];
  const float* betas  = (const float*)d_in[6];
  const float* lin1_w = (const float*)d_in[7];
  const float* lin1_b = (const float*)d_in[8];

  int N = in_sizes[0] / D;   // 50000 (multiple of 16, <= 65536 for the scan)
  int E = in_sizes[1];       // 800000
  long nd = (long)N * D;
  int nblk = (N + SCAN_B - 1) / SCAN_B;

  float* wf       = (float*)d_ws;
  float* norm_src = wf; wf += N;
  float* norm_dst = wf; wf += N;
  float* bnsum    = wf; wf += D;          // bnsum/bnsumsq contiguous (one zero)
  float* bnsumsq  = wf; wf += D;
  float* bnscale  = wf; wf += D;
  float* bnshift  = wf; wf += D;
  float* agg      = wf; wf += nd;         // gather output; GEMM runs in place
  float* hbuf     = wf; wf += nd;         // per-layer activations
  float* jk       = wf; wf += nd;         // JumpingKnowledge running max
  int* wi         = (int*)wf;
  int* deg_out_i  = wi; wi += N;          // deg_out/deg_in/fill contiguous
  int* deg_in_i   = wi; wi += N;
  int* fill       = wi; wi += N;
  int* row_ptr    = wi; wi += N + 1;
  int* col_idx    = wi; wi += E;
  int* blk_sum    = wi; wi += SCAN_B;
  int* blk_off    = wi; wi += SCAN_B;
  float* outf     = (float*)d_out;

  // --- CSR build (once per call) -------------------------------------------
  zero_b32<<<256, 256, 0, stream>>>(deg_out_i, 3L * N);   // degs + fill
  degree_kernel<<<(E + 255) / 256, 256, 0, stream>>>(src, dst, deg_out_i, deg_in_i, E);
  norm_kernel<<<(N + 255) / 256, 256, 0, stream>>>(deg_out_i, norm_src, N);
  norm_kernel<<<(N + 255) / 256, 256, 0, stream>>>(deg_in_i, norm_dst, N);
  block_sum_kernel<<<nblk, SCAN_B, 0, stream>>>(deg_in_i, blk_sum, N);
  scan_offsets_kernel<<<1, SCAN_B, 0, stream>>>(blk_sum, blk_off, nblk);
  scan_write_kernel<<<nblk, SCAN_B, 0, stream>>>(deg_in_i, blk_off, row_ptr, N);
  csr_fill_kernel<<<(E + 255) / 256, 256, 0, stream>>>(src, dst, row_ptr, fill, col_idx, E);

  // --- layers --------------------------------------------------------------
  const float* h = x;
  for (int i = 0; i < NLAYERS; ++i) {
    zero_b32<<<1, 256, 0, stream>>>((int*)bnsum, 2 * D);
    gather_kernel<<<(N * 32 + 255) / 256, 256, 0, stream>>>(h, norm_src, norm_dst,
                                                            row_ptr, col_idx, agg, N);
    gemm_relu_kernel<<<N / 16, 256, 0, stream>>>(agg, Ws + (long)i * D * D, bs + i * D,
                                                 agg /*in place*/, bnsum, bnsumsq, N);
    bn_finalize<<<1, D, 0, stream>>>(bnsum, bnsumsq, gammas + i * D, betas + i * D,
                                     bnscale, bnshift, 1.f / (float)N);
    bn_apply<<<2048, 256, 0, stream>>>(agg, bnscale, bnshift, hbuf, jk, i == 0, nd);
    h = hbuf;
  }

  // --- final linear head ---------------------------------------------------
  gemm_relu_kernel<<<N / 16, 256, 0, stream>>>(jk, lin1_w, lin1_b, outf,
                                               nullptr, nullptr, N);
}